// CustomDecoupledAPPNP_2877628089022
// MI455X (gfx1250) — compile-verified
//
#include <hip/hip_runtime.h>

typedef __attribute__((ext_vector_type(2))) float v2f;
typedef __attribute__((ext_vector_type(8))) float v8f;

// pointee/pointer types matching the async-LDS builtin's parameters
typedef __attribute__((__vector_size__(4 * sizeof(int)))) int vi4;
typedef __attribute__((address_space(1))) vi4 as1_vi4;   // global
typedef __attribute__((address_space(3))) vi4 as3_vi4;   // LDS

#define N_NODES 100000
#define N_EDGES 800000
#define D_DATA 100
#define H_DIM 256
#define N_CLS 47
#define K_ITERS 10
#define ALPHA 0.1f

#define HSTRIDE 104   // floats; rows 16B-aligned for b128 async copy (2-way bank conflict worst case)
#define ZSTRIDE 257   // odd stride -> conflict-free column reads through LDS

#if __has_builtin(__builtin_amdgcn_global_load_async_to_lds_b128)
#define HAVE_ASYNC_LDS 1
#else
#define HAVE_ASYNC_LDS 0
#endif

__device__ __forceinline__ void async_wait_all() {
#if __has_builtin(__builtin_amdgcn_s_wait_asynccnt)
    __builtin_amdgcn_s_wait_asynccnt(0);
#elif HAVE_ASYNC_LDS
    asm volatile("s_wait_asynccnt 0x0" ::: "memory");
#endif
}

// ---------------------------------------------------------------- utilities
__global__ void zero_f32_kernel(float* __restrict__ p, int n) {
    int i = blockIdx.x * blockDim.x + threadIdx.x;
    if (i < n) p[i] = 0.0f;
}

__global__ void deg_kernel(const int* __restrict__ dst, float* __restrict__ deg) {
    int e = blockIdx.x * blockDim.x + threadIdx.x;
    if (e < N_EDGES) atomicAdd(&deg[dst[e]], 1.0f);
}

__global__ void norm_kernel(float* __restrict__ norm) {
    int n = blockIdx.x * blockDim.x + threadIdx.x;
    if (n < N_NODES) {
        float d = norm[n];
        norm[n] = rsqrtf(fmaxf(d, 1.0f));
    }
}

// ------------------------------------------------- propagation: edge scatter
// one wave32 per edge; lanes 0..24 each move a float4 chunk of the 100-float row
__global__ __launch_bounds__(256) void edge_kernel(
    const float* __restrict__ h, const float* __restrict__ norm,
    const int* __restrict__ src, const int* __restrict__ dst,
    float* __restrict__ agg) {
    int tid  = blockIdx.x * blockDim.x + threadIdx.x;
    int e    = tid >> 5;
    int lane = tid & 31;
    if (e >= N_EDGES) return;
    int s = src[e];
    int d = dst[e];
    float ns = norm[s];
    if (lane < D_DATA / 4) {
        const float4* hp = (const float4*)(h + (size_t)s * D_DATA);
        float4 v = hp[lane];
        float* ap = agg + (size_t)d * D_DATA + lane * 4;
        atomicAdd(ap + 0, v.x * ns);
        atomicAdd(ap + 1, v.y * ns);
        atomicAdd(ap + 2, v.z * ns);
        atomicAdd(ap + 3, v.w * ns);
    }
}

// ---------------------------------------- propagation: node update + teleport
// h_out = (1-a)*agg*norm + a*x ; also re-zeros agg for the next iteration
__global__ __launch_bounds__(256) void node_kernel(
    float* __restrict__ agg, const float* __restrict__ x,
    const float* __restrict__ norm, float* __restrict__ hout) {
    int i = blockIdx.x * blockDim.x + threadIdx.x;   // float4 index
    const int CHUNKS = D_DATA / 4;                   // 25 per node
    if (i >= N_NODES * CHUNKS) return;
    int n = i / CHUNKS;
    float nm = norm[n] * (1.0f - ALPHA);
    float4 a  = ((float4*)agg)[i];
    float4 xv = ((const float4*)x)[i];
    float4 r;
    r.x = a.x * nm + ALPHA * xv.x;
    r.y = a.y * nm + ALPHA * xv.y;
    r.z = a.z * nm + ALPHA * xv.z;
    r.w = a.w * nm + ALPHA * xv.w;
    ((float4*)hout)[i] = r;
    float4 z4; z4.x = 0.f; z4.y = 0.f; z4.z = 0.f; z4.w = 0.f;
    ((float4*)agg)[i] = z4;
}

// -------------------------------------------------------- fused MLP (WMMA)
// block = 256 threads (8 wave32) handles 32 node rows (two 16-row m-tiles).
// Stage h[32,100] into LDS (async global->LDS when available).
// GEMM1: wave w -> m-tile (w>>2), hidden quarter (w&3): 4 N-tiles x 25 K-steps
//        of v_wmma_f32_16x16x4_f32; bias+relu into padded LDS z-tile.
// GEMM2: waves 0..5 -> (m-tile, n-tile) pairs over K=256; mask pad column 47.
__global__ __launch_bounds__(256) void mlp_kernel(
    const float* __restrict__ h,  const float* __restrict__ W1,
    const float* __restrict__ b1, const float* __restrict__ W2,
    const float* __restrict__ b2, float* __restrict__ out) {
    __shared__ float hsh[2][16 * HSTRIDE];
    __shared__ float zsh[2][16 * ZSTRIDE];

    const int tid    = threadIdx.x;
    const int lane   = tid & 31;
    const int wave   = tid >> 5;
    const int lanelo = lane & 15;     // m for A-frags, n for B/C-frags
    const int khalf  = lane >> 4;     // selects K pair {0,1} vs {2,3}
    const int m0     = blockIdx.x * 32;

    // ---- stage h tile: 32 rows x 25 float4 chunks ----
    for (int c = tid; c < 32 * (D_DATA / 4); c += 256) {
        int row = c / (D_DATA / 4);
        int ch  = c % (D_DATA / 4);
        const float* gp = h + (size_t)(m0 + row) * D_DATA + ch * 4;
        float* lp = &hsh[row >> 4][(row & 15) * HSTRIDE + ch * 4];
#if HAVE_ASYNC_LDS
        __builtin_amdgcn_global_load_async_to_lds_b128(
            (as1_vi4*)(void*)gp, (as3_vi4*)(void*)lp, 0, 0);
#else
        *(float4*)lp = *(const float4*)gp;
#endif
    }
    async_wait_all();
    __syncthreads();

    // ---- GEMM1 ----
    const int mt = wave >> 2;   // which 16-row m-tile
    const int q  = wave & 3;    // hidden-dim quarter [64q, 64q+64)
    v8f acc[4] = {};
#pragma unroll
    for (int k0 = 0; k0 < D_DATA; k0 += 4) {
        // A fragment: lane -> hsh[mt][lanelo][k0 + 2*khalf + {0,1}]  (8B aligned)
        v2f a = *(const v2f*)&hsh[mt][lanelo * HSTRIDE + k0 + 2 * khalf];
#pragma unroll
        for (int t = 0; t < 4; ++t) {
            int ncol = q * 64 + t * 16 + lanelo;
            const float* bcol = W1 + (size_t)(k0 + 2 * khalf) * H_DIM + ncol;
            v2f b;
            b.x = bcol[0];
            b.y = bcol[H_DIM];
            acc[t] = __builtin_amdgcn_wmma_f32_16x16x4_f32(
                false, a, false, b, (short)0, acc[t], false, false);
        }
    }
    // bias + relu, C-layout -> LDS row-major z tile
#pragma unroll
    for (int t = 0; t < 4; ++t) {
        int n = q * 64 + t * 16 + lanelo;
        float bias = b1[n];
#pragma unroll
        for (int v = 0; v < 8; ++v) {
            int m = v + 8 * khalf;
            float val = acc[t][v] + bias;
            zsh[mt][m * ZSTRIDE + n] = val > 0.0f ? val : 0.0f;
        }
    }
    __syncthreads();

    // ---- GEMM2 ---- (waves 0..5; wave-uniform branch so EXEC stays full)
    if (wave < 6) {
        const int mt2 = wave / 3;
        const int nt  = wave % 3;
        v8f c = {};
        int n = nt * 16 + lanelo;
        bool nok = (n < N_CLS);
#pragma unroll 8
        for (int k0 = 0; k0 < H_DIM; k0 += 4) {
            int krow = k0 + 2 * khalf;
            v2f a;
            a.x = zsh[mt2][lanelo * ZSTRIDE + krow];
            a.y = zsh[mt2][lanelo * ZSTRIDE + krow + 1];
            v2f b;
            b.x = nok ? W2[(size_t)krow * N_CLS + n] : 0.0f;
            b.y = nok ? W2[(size_t)(krow + 1) * N_CLS + n] : 0.0f;
            c = __builtin_amdgcn_wmma_f32_16x16x4_f32(
                false, a, false, b, (short)0, c, false, false);
        }
        if (nok) {
            float bias = b2[n];
#pragma unroll
            for (int v = 0; v < 8; ++v) {
                int m = v + 8 * khalf;
                out[(size_t)(m0 + mt2 * 16 + m) * N_CLS + n] = c[v] + bias;
            }
        }
    }
}

// ------------------------------------------------------------------ driver
extern "C" void kernel_launch(void* const* d_in, const int* in_sizes, int n_in,
                              void* d_out, int out_size, void* d_ws, size_t ws_size,
                              hipStream_t stream) {
    const float* x   = (const float*)d_in[0];
    const int*   src = (const int*)  d_in[1];
    const int*   dst = (const int*)  d_in[2];
    const float* W1  = (const float*)d_in[3];
    const float* b1  = (const float*)d_in[4];
    const float* W2  = (const float*)d_in[5];
    const float* b2  = (const float*)d_in[6];
    float* out = (float*)d_out;

    // workspace layout (floats): norm | hA | hB | agg  (~120.5 MB total)
    float* ws   = (float*)d_ws;
    float* norm = ws;
    float* hA   = norm + N_NODES;
    float* hB   = hA + (size_t)N_NODES * D_DATA;
    float* agg  = hB + (size_t)N_NODES * D_DATA;

    const int T = 256;
    const int nd = N_NODES * D_DATA;

    // degrees -> norm; zero agg
    zero_f32_kernel<<<(N_NODES + T - 1) / T, T, 0, stream>>>(norm, N_NODES);
    zero_f32_kernel<<<(nd + T - 1) / T, T, 0, stream>>>(agg, nd);
    deg_kernel<<<(N_EDGES + T - 1) / T, T, 0, stream>>>(dst, norm);
    norm_kernel<<<(N_NODES + T - 1) / T, T, 0, stream>>>(norm);

    // K propagation rounds (ping-pong h; agg re-zeroed inside node_kernel)
    const int edge_blocks = (N_EDGES * 32 + T - 1) / T;
    const int node_elems  = N_NODES * (D_DATA / 4);
    const int node_blocks = (node_elems + T - 1) / T;
    const float* hin = x;
    float* cur = hA;
    float* nxt = hB;
    for (int it = 0; it < K_ITERS; ++it) {
        edge_kernel<<<edge_blocks, T, 0, stream>>>(hin, norm, src, dst, agg);
        node_kernel<<<node_blocks, T, 0, stream>>>(agg, x, norm, cur);
        hin = cur;
        float* tmp = cur; cur = nxt; nxt = tmp;
    }

    // fused MLP with fp32 WMMA (32 rows per 8-wave block)
    mlp_kernel<<<N_NODES / 32, 256, 0, stream>>>(hin, W1, b1, W2, b2, out);
}